// Generator_11192684773525
// MI455X (gfx1250) — compile-verified
//
#include <hip/hip_runtime.h>
#include <cmath>

#define BDIM 128
#define SDIM 64
#define EDIM 256
#define HDIM 256
#define DDIM 256
#define ADIM 256
#define VDIM 10000

typedef float v2f __attribute__((ext_vector_type(2)));
typedef float v8f __attribute__((ext_vector_type(8)));

// =====================================================================
// Register-blocked fp32 WMMA GEMM:
//   C[M,N] = A[M,K] @ W[N,K]^T + bias   (opt tanh epilogue)
// One wave owns an MT x NT super-tile of 16x16 tiles; A-fragments are
// reused across NT WMMAs and B-fragments across MT WMMAs, cutting
// global loads per V_WMMA_F32_16X16X4_F32 from 2 to (MT+NT)/(MT*NT).
// Fragment layout (fp32 16x16x4):
//   A frag: lanes 0-15 hold M=0..15 @ K={k,k+1}; lanes 16-31 @ K={k+2,k+3}
//   B frag: same addressing with N in place of M (W is row-major (N,K))
//   C/D:    lanes 0-15 rows 0..7 in v[0..7] at col=lane; lanes 16-31 rows 8..15
// =====================================================================
template <int MT, int NT, int ACT>
__global__ __launch_bounds__(256) void wmma_gemm_blk(
    const float* __restrict__ A, int lda,
    const float* __restrict__ W, int ldw,
    const float* __restrict__ bias,
    float* __restrict__ C, int ldc,
    int M, int N, int K)
{
    const int mgrp  = (M >> 4) / MT;
    const int ngrp  = (N >> 4) / NT;
    const int total = mgrp * ngrp;
    const int wave  = blockIdx.x * 8 + (threadIdx.x >> 5);
    if (wave >= total) return;           // whole-wave exit: EXEC stays all-1 for WMMA
    const int mg   = wave / ngrp;
    const int ng   = wave - mg * ngrp;
    const int lane = threadIdx.x & 31;
    const int l16  = lane & 15;
    const int half = lane >> 4;

    const float* a_row[MT];
    const float* w_row[NT];
    #pragma unroll
    for (int i = 0; i < MT; ++i)
        a_row[i] = A + (size_t)((mg * MT + i) * 16 + l16) * lda + half * 2;
    #pragma unroll
    for (int j = 0; j < NT; ++j)
        w_row[j] = W + (size_t)((ng * NT + j) * 16 + l16) * ldw + half * 2;

    const v8f zero = {};
    v8f acc[MT][NT];
    #pragma unroll
    for (int i = 0; i < MT; ++i)
        #pragma unroll
        for (int j = 0; j < NT; ++j) acc[i][j] = zero;

    #pragma unroll 4
    for (int k = 0; k < K; k += 4) {
        v2f av[MT], bv[NT];
        #pragma unroll
        for (int i = 0; i < MT; ++i) av[i] = *(const v2f*)(a_row[i] + k);
        #pragma unroll
        for (int j = 0; j < NT; ++j) bv[j] = *(const v2f*)(w_row[j] + k);
        #pragma unroll
        for (int i = 0; i < MT; ++i)
            #pragma unroll
            for (int j = 0; j < NT; ++j)
                acc[i][j] = __builtin_amdgcn_wmma_f32_16x16x4_f32(
                    false, av[i], false, bv[j], (short)0, acc[i][j], false, false);
    }

    #pragma unroll
    for (int j = 0; j < NT; ++j) {
        const int col = (ng * NT + j) * 16 + l16;
        const float bb = bias ? bias[col] : 0.0f;
        #pragma unroll
        for (int i = 0; i < MT; ++i) {
            #pragma unroll
            for (int r = 0; r < 8; ++r) {
                int row = (mg * MT + i) * 16 + half * 8 + r;
                float v = acc[i][j][r] + bb;
                if (ACT == 1) v = tanhf(v);
                C[(size_t)row * ldc + col] = v;
            }
        }
    }
}

// ---------------------------------------------------------------------
__global__ void fillz(float* __restrict__ p, int n) {
    int i = blockIdx.x * blockDim.x + threadIdx.x;
    if (i < n) p[i] = 0.0f;
}

// emb[b,s,:] = emb_table[tok[b,s]] * mask[b,s]
__global__ __launch_bounds__(256) void embed_all(
    const int* __restrict__ inp, const float* __restrict__ table,
    float* __restrict__ emb)
{
    int bs = blockIdx.x;
    int j  = threadIdx.x;
    int tok  = inp[bs * 2 + 0];
    float mk = (float)inp[bs * 2 + 1];
    emb[(size_t)bs * EDIM + j] = table[(size_t)tok * EDIM + j] * mk;
}

// GRU cell: h' = (1-z)*n + z*h, in-place on h; optional second write (fused concat)
__global__ __launch_bounds__(256) void gru_combine(
    const float* __restrict__ xp, int xp_bstride,
    const float* __restrict__ hh,
    float* __restrict__ h,
    float* __restrict__ out_slice, int out_bstride)
{
    int b = blockIdx.x;
    int j = threadIdx.x;
    const float* xpr = xp + (size_t)b * xp_bstride;
    const float* hhr = hh + (size_t)b * (3 * HDIM);
    float rx = xpr[j], zx = xpr[HDIM + j], nx = xpr[2 * HDIM + j];
    float rh = hhr[j], zh = hhr[HDIM + j], nh = hhr[2 * HDIM + j];
    float r = 1.0f / (1.0f + expf(-(rx + rh)));
    float z = 1.0f / (1.0f + expf(-(zx + zh)));
    float n = tanhf(nx + r * nh);
    float hv = h[(size_t)b * HDIM + j];
    float hn = (1.0f - z) * n + z * hv;
    h[(size_t)b * HDIM + j] = hn;
    if (out_slice) out_slice[(size_t)b * out_bstride + j] = hn;
}

__global__ __launch_bounds__(256) void copy_rows(
    const float* __restrict__ src, int sstride,
    float* __restrict__ dst, int dstride)
{
    int b = blockIdx.x;
    int j = threadIdx.x;
    dst[(size_t)b * dstride + j] = src[(size_t)b * sstride + j];
}

// esum[b,s] = sum_a tanh(base[b,s,a] + hWh[b,a])
__global__ __launch_bounds__(256) void attn_energy(
    const float* __restrict__ base, const float* __restrict__ hWh,
    float* __restrict__ esum)
{
    int bs = blockIdx.x;
    int b  = bs >> 6;
    int a  = threadIdx.x;
    float v = tanhf(base[(size_t)bs * ADIM + a] + hWh[(size_t)b * ADIM + a]);
    __shared__ float sh[256];
    sh[a] = v; __syncthreads();
    for (int off = 128; off > 0; off >>= 1) {
        if (a < off) sh[a] += sh[a + off];
        __syncthreads();
    }
    if (a == 0) esum[bs] = sh[0];
}

// softmax over S=64 per batch row, in-place
__global__ void attn_softmax(float* __restrict__ esum) {
    int b = blockIdx.x;
    int s = threadIdx.x;
    __shared__ float sh[SDIM];
    __shared__ float mxs, sms;
    float v = esum[b * SDIM + s];
    sh[s] = v; __syncthreads();
    if (s == 0) { float m = sh[0]; for (int i = 1; i < SDIM; ++i) m = fmaxf(m, sh[i]); mxs = m; }
    __syncthreads();
    float ex = expf(v - mxs);
    sh[s] = ex; __syncthreads();
    if (s == 0) { float t = 0.f; for (int i = 0; i < SDIM; ++i) t += sh[i]; sms = t; }
    __syncthreads();
    esum[b * SDIM + s] = ex / sms;
}

// w[b,:] = sum_s a[b,s] * enc_out[b,s,:]; fused into rnn_in[:,256:768] and X[:,256:768]
__global__ __launch_bounds__(512) void attn_wsum(
    const float* __restrict__ attw, const float* __restrict__ enc_out,
    float* __restrict__ rnn_in, float* __restrict__ X)
{
    int b = blockIdx.x;
    int j = threadIdx.x;   // 512 = 2H
    float acc = 0.f;
    #pragma unroll 4
    for (int s = 0; s < SDIM; ++s)
        acc += attw[b * SDIM + s] * enc_out[((size_t)b * SDIM + s) * (2 * HDIM) + j];
    rnn_in[(size_t)b * 768 + EDIM + j]  = acc;
    X[(size_t)b * 1024 + DDIM + j]      = acc;
}

// e = emb_table[tok]; fused into rnn_in[:,0:256] and X[:,768:1024]
__global__ __launch_bounds__(256) void gather_e(
    const int* __restrict__ tok, const float* __restrict__ table,
    float* __restrict__ rnn_in, float* __restrict__ X)
{
    int b = blockIdx.x;
    int j = threadIdx.x;
    float v = table[(size_t)tok[b] * EDIM + j];
    rnn_in[(size_t)b * 768 + j]        = v;
    X[(size_t)b * 1024 + 768 + j]      = v;
}

__global__ __launch_bounds__(256) void fill_onehot(float* __restrict__ logits) {
    int b = blockIdx.x;
    for (int v = threadIdx.x; v < VDIM; v += 256)
        logits[(size_t)b * VDIM + v] = (v == 0) ? 1.0f : 0.0f;
}

// out[b*S+t, :] = log_softmax(logits[b,:]); tok[b] = first argmax
__global__ __launch_bounds__(256) void logsoftmax_argmax(
    const float* __restrict__ logits, float* __restrict__ out,
    int t, int* __restrict__ tok)
{
    int b = blockIdx.x;
    int tid = threadIdx.x;
    const float* row = logits + (size_t)b * VDIM;
    __shared__ float sv[256];
    __shared__ int   si[256];
    float mx = -3.4e38f; int mi = 0x7fffffff;
    for (int v = tid; v < VDIM; v += 256) {
        float x = row[v];
        if (x > mx) { mx = x; mi = v; }
    }
    sv[tid] = mx; si[tid] = mi; __syncthreads();
    for (int off = 128; off > 0; off >>= 1) {
        if (tid < off) {
            float v2 = sv[tid + off]; int i2 = si[tid + off];
            if (v2 > sv[tid] || (v2 == sv[tid] && i2 < si[tid])) { sv[tid] = v2; si[tid] = i2; }
        }
        __syncthreads();
    }
    const float MX = sv[0];
    const int   MI = si[0];
    __syncthreads();
    float part = 0.f;
    for (int v = tid; v < VDIM; v += 256) part += expf(row[v] - MX);
    sv[tid] = part; __syncthreads();
    for (int off = 128; off > 0; off >>= 1) {
        if (tid < off) sv[tid] += sv[tid + off];
        __syncthreads();
    }
    const float LSE = MX + logf(sv[0]);
    float* orow = out + ((size_t)b * SDIM + t) * VDIM;
    for (int v = tid; v < VDIM; v += 256) orow[v] = row[v] - LSE;
    if (tid == 0) tok[b] = MI;
}

// =====================================================================
static void gemm(hipStream_t st, const float* A, int lda, const float* W, int ldw,
                 const float* bias, float* C, int ldc, int M, int N, int K, bool act)
{
    const int mt = M / 16, nt = N / 16;
#define LAUNCH_GEMM(MT, NT)                                                          \
    do {                                                                             \
        int total  = (mt / MT) * (nt / NT);                                          \
        int blocks = (total + 7) / 8;                                                \
        if (act) wmma_gemm_blk<MT, NT, 1><<<blocks, 256, 0, st>>>(A, lda, W, ldw,    \
                                                  bias, C, ldc, M, N, K);            \
        else     wmma_gemm_blk<MT, NT, 0><<<blocks, 256, 0, st>>>(A, lda, W, ldw,    \
                                                  bias, C, ldc, M, N, K);            \
    } while (0)

    if ((mt % 2) == 0 && (nt % 4) == 0)      LAUNCH_GEMM(2, 4);  // N = 256, 768
    else if ((mt % 2) == 0 && (nt % 5) == 0) LAUNCH_GEMM(2, 5);  // N = 10000 (625 = 5^4)
    else                                     LAUNCH_GEMM(1, 1);  // generic fallback
#undef LAUNCH_GEMM
}

extern "C" void kernel_launch(void* const* d_in, const int* in_sizes, int n_in,
                              void* d_out, int out_size, void* d_ws, size_t ws_size,
                              hipStream_t stream)
{
    (void)in_sizes; (void)n_in; (void)out_size; (void)ws_size;
    const int*   inp       = (const int*)  d_in[0];
    const float* emb_table = (const float*)d_in[1];
    const float* Wih_f     = (const float*)d_in[2];
    const float* Whh_f     = (const float*)d_in[3];
    const float* bih_f     = (const float*)d_in[4];
    const float* bhh_f     = (const float*)d_in[5];
    const float* Wih_b     = (const float*)d_in[6];
    const float* Whh_b     = (const float*)d_in[7];
    const float* bih_b     = (const float*)d_in[8];
    const float* bhh_b     = (const float*)d_in[9];
    const float* fc_W      = (const float*)d_in[10];
    const float* fc_b      = (const float*)d_in[11];
    const float* attn_W    = (const float*)d_in[12];
    const float* attn_b    = (const float*)d_in[13];
    const float* dec_Wih   = (const float*)d_in[14];
    const float* dec_Whh   = (const float*)d_in[15];
    const float* dec_bih   = (const float*)d_in[16];
    const float* dec_bhh   = (const float*)d_in[17];
    const float* out_W     = (const float*)d_in[18];
    const float* out_b     = (const float*)d_in[19];
    float* out = (float*)d_out;

    // ---- workspace layout (floats) ----
    float* ws = (float*)d_ws;
    size_t off = 0;
    float* emb     = ws + off; off += (size_t)BDIM * SDIM * EDIM;       // 2.10M
    float* xp_f    = ws + off; off += (size_t)BDIM * SDIM * 3 * HDIM;   // 6.29M
    float* xp_b    = ws + off; off += (size_t)BDIM * SDIM * 3 * HDIM;   // 6.29M
    float* enc_out = ws + off; off += (size_t)BDIM * SDIM * 2 * HDIM;   // 4.19M
    float* base    = ws + off; off += (size_t)BDIM * SDIM * ADIM;       // 2.10M
    float* h       = ws + off; off += (size_t)BDIM * HDIM;
    float* hh      = ws + off; off += (size_t)BDIM * 3 * HDIM;
    float* henc    = ws + off; off += (size_t)BDIM * 2 * HDIM;
    float* hWh     = ws + off; off += (size_t)BDIM * ADIM;
    float* esum    = ws + off; off += (size_t)BDIM * SDIM;
    float* rnn_in  = ws + off; off += (size_t)BDIM * 768;
    float* xp_dec  = ws + off; off += (size_t)BDIM * 768;
    float* X       = ws + off; off += (size_t)BDIM * 1024;
    float* logits  = ws + off; off += (size_t)BDIM * VDIM;
    int*   tok     = (int*)(ws + off);                                  // 128 ints

    // ================= Encoder =================
    embed_all<<<BDIM * SDIM, 256, 0, stream>>>(inp, emb_table, emb);

    // batched input projections for both directions
    gemm(stream, emb, EDIM, Wih_f, EDIM, bih_f, xp_f, 3 * HDIM, BDIM * SDIM, 3 * HDIM, EDIM, false);
    gemm(stream, emb, EDIM, Wih_b, EDIM, bih_b, xp_b, 3 * HDIM, BDIM * SDIM, 3 * HDIM, EDIM, false);

    const int hN = BDIM * HDIM;
    // forward scan -> enc_out[:, s, 0:256]
    fillz<<<(hN + 255) / 256, 256, 0, stream>>>(h, hN);
    for (int s = 0; s < SDIM; ++s) {
        gemm(stream, h, HDIM, Whh_f, HDIM, bhh_f, hh, 3 * HDIM, BDIM, 3 * HDIM, HDIM, false);
        gru_combine<<<BDIM, 256, 0, stream>>>(xp_f + (size_t)s * 3 * HDIM, SDIM * 3 * HDIM,
                                              hh, h,
                                              enc_out + (size_t)s * 2 * HDIM, SDIM * 2 * HDIM);
    }
    copy_rows<<<BDIM, 256, 0, stream>>>(h, HDIM, henc, 2 * HDIM);          // f_h

    // backward scan -> enc_out[:, s, 256:512]
    fillz<<<(hN + 255) / 256, 256, 0, stream>>>(h, hN);
    for (int s = SDIM - 1; s >= 0; --s) {
        gemm(stream, h, HDIM, Whh_b, HDIM, bhh_b, hh, 3 * HDIM, BDIM, 3 * HDIM, HDIM, false);
        gru_combine<<<BDIM, 256, 0, stream>>>(xp_b + (size_t)s * 3 * HDIM, SDIM * 3 * HDIM,
                                              hh, h,
                                              enc_out + (size_t)s * 2 * HDIM + HDIM, SDIM * 2 * HDIM);
    }
    copy_rows<<<BDIM, 256, 0, stream>>>(h, HDIM, henc + HDIM, 2 * HDIM);   // b_h

    // hidden = tanh([f_h,b_h] @ fc_W^T + fc_b)  -> h (decoder state)
    gemm(stream, henc, 2 * HDIM, fc_W, 2 * HDIM, fc_b, h, DDIM, BDIM, DDIM, 2 * HDIM, true);

    // attention precompute: base = enc_out @ attn_W[:,256:]^T + attn_b
    gemm(stream, enc_out, 2 * HDIM, attn_W + DDIM, 2 * HDIM + DDIM, attn_b,
         base, ADIM, BDIM * SDIM, ADIM, 2 * HDIM, false);

    // ================= Decoder =================
    // t = 0: one-hot(0) rows through the same log-softmax (argmax -> tok = 0 = tok0)
    fill_onehot<<<BDIM, 256, 0, stream>>>(logits);
    logsoftmax_argmax<<<BDIM, 256, 0, stream>>>(logits, out, 0, tok);

    for (int t = 1; t < SDIM; ++t) {
        // attention with pre-update h
        gemm(stream, h, DDIM, attn_W, 2 * HDIM + DDIM, nullptr, hWh, ADIM, BDIM, ADIM, DDIM, false);
        attn_energy<<<BDIM * SDIM, 256, 0, stream>>>(base, hWh, esum);
        attn_softmax<<<BDIM, SDIM, 0, stream>>>(esum);
        attn_wsum<<<BDIM, 512, 0, stream>>>(esum, enc_out, rnn_in, X);     // w -> rnn_in[:,256:], X[:,256:768]
        gather_e<<<BDIM, 256, 0, stream>>>(tok, emb_table, rnn_in, X);     // e -> rnn_in[:,:256], X[:,768:]

        // GRU cell
        gemm(stream, rnn_in, 768, dec_Wih, 768, dec_bih, xp_dec, 3 * DDIM, BDIM, 3 * DDIM, 768, false);
        gemm(stream, h, DDIM, dec_Whh, DDIM, dec_bhh, hh, 3 * DDIM, BDIM, 3 * DDIM, DDIM, false);
        gru_combine<<<BDIM, 256, 0, stream>>>(xp_dec, 3 * DDIM, hh, h, X, 1024); // h_new -> X[:,:256]

        // output projection (the 165-GFLOP hot loop; out_W L2-resident)
        gemm(stream, X, 1024, out_W, 1024, out_b, logits, VDIM, BDIM, VDIM, 1024, false);
        logsoftmax_argmax<<<BDIM, 256, 0, stream>>>(logits, out, t, tok);
    }
}